// GCN_50663434224293
// MI455X (gfx1250) — compile-verified
//
#include <hip/hip_runtime.h>
#include <hip/hip_bf16.h>

typedef __attribute__((ext_vector_type(2))) float v2f;
typedef __attribute__((ext_vector_type(8))) float v8f;
typedef __attribute__((ext_vector_type(4))) int   v4i;

#define IN_C  512
#define HID_C 256
#define OUT_C 64

// ---- gfx1250 async memory->LDS path (guarded; falls back to sync copy) ----
#if defined(__has_builtin)
#  if __has_builtin(__builtin_amdgcn_global_load_async_to_lds_b128)
#    define HAVE_ASYNC_LDS 1
#  endif
#  if __has_builtin(__builtin_amdgcn_s_wait_asynccnt)
#    define HAVE_WAIT_ASYNC 1
#  endif
#endif
#ifndef HAVE_ASYNC_LDS
#  define HAVE_ASYNC_LDS 0
#endif
#ifndef HAVE_WAIT_ASYNC
#  define HAVE_WAIT_ASYNC 0
#endif

#if HAVE_ASYNC_LDS
typedef __attribute__((address_space(1))) v4i gas_v4i;  // global AS pointer elem
typedef __attribute__((address_space(3))) v4i las_v4i;  // LDS AS pointer elem
#endif

__device__ __forceinline__ void copy16_to_lds(const float* gsrc, float* ldst) {
#if HAVE_ASYNC_LDS
    __builtin_amdgcn_global_load_async_to_lds_b128(
        (gas_v4i*)gsrc, (las_v4i*)ldst, 0, 0);
#else
    *(float4*)ldst = *(const float4*)gsrc;
#endif
}

__device__ __forceinline__ void wait_lds_copies() {
#if HAVE_ASYNC_LDS
#  if HAVE_WAIT_ASYNC
    __builtin_amdgcn_s_wait_asynccnt(0);
#  else
    asm volatile("s_wait_asynccnt 0" ::: "memory");
#  endif
#endif
}

// ---------------------------------------------------------------------------
// Zero-fill (graph-safe init of accumulators)
// ---------------------------------------------------------------------------
__global__ void zerok(float* __restrict__ p, size_t n) {
    size_t i = (size_t)blockIdx.x * blockDim.x + threadIdx.x;
    size_t stride = (size_t)gridDim.x * blockDim.x;
    for (; i < n; i += stride) p[i] = 0.0f;
}

// ---------------------------------------------------------------------------
// GEMM1: h1[N,256] = x[N,512] @ W1[512,256] + b1
// Block = 256 threads (8 waves). Block computes 32 rows x 256 cols
// (32 rows halves per-block W1 re-read vs 16). Wave w owns cols [32w,32w+32)
// as 2 N-tiles x 2 M-tiles (B fragments reused across M-tiles).
// A staged in LDS, stride 516 floats (516 % 64 == 4 -> conflict-free rows;
// 516*4 bytes = 16B-multiple -> b128 async copies stay aligned).
// ---------------------------------------------------------------------------
__global__ __launch_bounds__(256) void gemm1(const float* __restrict__ x,
                                             const float* __restrict__ W1,
                                             const float* __restrict__ b1,
                                             float* __restrict__ h1, int N) {
    __shared__ float sx[32 * 516];   // 66 KB (gfx1250 WGP LDS = 320 KB)
    const int tid = threadIdx.x;
    const int wave = tid >> 5, lane = tid & 31;
    const int rowBase = blockIdx.x * 32;

    // stage: 32 rows x 512 cols = 4096 float4 slots, 16 per thread
    for (int i = tid; i < 32 * 128; i += 256) {
        const int r = i >> 7, kq = i & 127;
        const int gr = rowBase + r;
        float* p = &sx[r * 516 + kq * 4];
        if (gr < N) {
            copy16_to_lds(x + (size_t)gr * IN_C + kq * 4, p);
        } else {
            p[0] = 0.f; p[1] = 0.f; p[2] = 0.f; p[3] = 0.f;
        }
    }
    // speculative prefetch of the next block's x tile (dropped if OOB)
    __builtin_prefetch(x + (size_t)(rowBase + 32) * IN_C + tid * 16, 0, 1);
    wait_lds_copies();
    __syncthreads();

    const int colA = wave * 32 + (lane & 15);
    const int colB = colA + 16;
    const int khalf = (lane >> 4) * 2;          // ISA 16x4 f32 A layout
    const int arow  = lane & 15;

    v8f acc00 = {}; v8f acc01 = {};             // M-tile0 x {colA,colB}
    v8f acc10 = {}; v8f acc11 = {};             // M-tile1 x {colA,colB}
    for (int k = 0; k < IN_C; k += 4) {
        const int ko = k + khalf;
        v2f a0, a1;
        a0.x = sx[arow * 516 + ko];
        a0.y = sx[arow * 516 + ko + 1];
        a1.x = sx[(arow + 16) * 516 + ko];
        a1.y = sx[(arow + 16) * 516 + ko + 1];
        v2f bA, bB;
        const size_t r0 = (size_t)ko * HID_C;
        const size_t r1 = r0 + HID_C;
        bA.x = W1[r0 + colA]; bA.y = W1[r1 + colA];
        bB.x = W1[r0 + colB]; bB.y = W1[r1 + colB];
        acc00 = __builtin_amdgcn_wmma_f32_16x16x4_f32(false, a0, false, bA,
                                                      (short)0, acc00, false, false);
        acc01 = __builtin_amdgcn_wmma_f32_16x16x4_f32(false, a0, false, bB,
                                                      (short)0, acc01, false, false);
        acc10 = __builtin_amdgcn_wmma_f32_16x16x4_f32(false, a1, false, bA,
                                                      (short)0, acc10, false, false);
        acc11 = __builtin_amdgcn_wmma_f32_16x16x4_f32(false, a1, false, bB,
                                                      (short)0, acc11, false, false);
    }

    // C/D layout: VGPR i holds row (i + 8*(lane>>4)), col = lane&15
    const int half = lane >> 4, c = lane & 15;
    const float biasA = b1[wave * 32 + c];
    const float biasB = b1[wave * 32 + 16 + c];
#pragma unroll
    for (int i = 0; i < 8; ++i) {
        const int r0 = rowBase + i + 8 * half;
        const int r1 = r0 + 16;
        if (r0 < N) {
            h1[(size_t)r0 * HID_C + wave * 32 + c]      = acc00[i] + biasA;
            h1[(size_t)r0 * HID_C + wave * 32 + 16 + c] = acc01[i] + biasB;
        }
        if (r1 < N) {
            h1[(size_t)r1 * HID_C + wave * 32 + c]      = acc10[i] + biasA;
            h1[(size_t)r1 * HID_C + wave * 32 + 16 + c] = acc11[i] + biasB;
        }
    }
}

// ---------------------------------------------------------------------------
// SpMM scatter, 256 channels: one wave per edge, float4 gather + atomic adds.
// acc[row[e], :] += w[e] * h[col[e], :]   (L2-resident gather + L2 atomics)
// ---------------------------------------------------------------------------
__global__ __launch_bounds__(256) void spmm256(const int* __restrict__ erow,
                                               const int* __restrict__ ecol,
                                               const float* __restrict__ ew,
                                               const float* __restrict__ h,
                                               float* __restrict__ acc, int E) {
    const int gw = (int)((blockIdx.x * (size_t)blockDim.x + threadIdx.x) >> 5);
    if (gw >= E) return;
    const int lane = threadIdx.x & 31;
    const int r = erow[gw], c = ecol[gw];
    const float w = ew[gw];
    const float* __restrict__ src = h + (size_t)c * HID_C;
    float* __restrict__ dst = acc + (size_t)r * HID_C;
#pragma unroll
    for (int ch = lane * 4; ch < HID_C; ch += 128) {
        const float4 v = *(const float4*)(src + ch);
        atomicAdd(dst + ch + 0, w * v.x);
        atomicAdd(dst + ch + 1, w * v.y);
        atomicAdd(dst + ch + 2, w * v.z);
        atomicAdd(dst + ch + 3, w * v.w);
    }
}

// ---------------------------------------------------------------------------
// GEMM2: h2[N,64] = relu(g[N,256]) @ W2[256,64] + b2
// Block = 128 threads (4 waves), 16 rows. Staging is a pure (async) copy;
// ReLU applied on the LDS->fragment read. LDS stride 260 (260 % 64 == 4).
// ---------------------------------------------------------------------------
__global__ __launch_bounds__(128) void gemm2(const float* __restrict__ g,
                                             const float* __restrict__ W2,
                                             const float* __restrict__ b2,
                                             float* __restrict__ h2, int N) {
    __shared__ float sa[16 * 260];
    const int tid = threadIdx.x;
    const int wave = tid >> 5, lane = tid & 31;
    const int rowBase = blockIdx.x * 16;

    // stage 16 rows x 256 cols = 1024 float4 slots, 8 per thread
    for (int i = tid; i < 16 * 64; i += 128) {
        const int r = i >> 6, kq = i & 63;
        const int gr = rowBase + r;
        float* p = &sa[r * 260 + kq * 4];
        if (gr < N) {
            copy16_to_lds(g + (size_t)gr * HID_C + kq * 4, p);
        } else {
            p[0] = 0.f; p[1] = 0.f; p[2] = 0.f; p[3] = 0.f;
        }
    }
    wait_lds_copies();
    __syncthreads();

    const int col   = wave * 16 + (lane & 15);
    const int khalf = (lane >> 4) * 2;
    const int arow  = lane & 15;

    v8f acc = {};
    for (int k = 0; k < HID_C; k += 4) {
        const int ko = k + khalf;
        v2f a;
        a.x = fmaxf(sa[arow * 260 + ko], 0.f);      // fused ReLU
        a.y = fmaxf(sa[arow * 260 + ko + 1], 0.f);
        v2f b;
        b.x = W2[(size_t)ko * OUT_C + col];
        b.y = W2[(size_t)(ko + 1) * OUT_C + col];
        acc = __builtin_amdgcn_wmma_f32_16x16x4_f32(false, a, false, b,
                                                    (short)0, acc, false, false);
    }

    const int half = lane >> 4, c = lane & 15;
    const float bias = b2[wave * 16 + c];
#pragma unroll
    for (int i = 0; i < 8; ++i) {
        const int r = rowBase + i + 8 * half;
        if (r < N) h2[(size_t)r * OUT_C + wave * 16 + c] = acc[i] + bias;
    }
}

// ---------------------------------------------------------------------------
// SpMM scatter, 64 channels: one wave per edge, float2 per lane.
// ---------------------------------------------------------------------------
__global__ __launch_bounds__(256) void spmm64(const int* __restrict__ erow,
                                              const int* __restrict__ ecol,
                                              const float* __restrict__ ew,
                                              const float* __restrict__ h,
                                              float* __restrict__ acc, int E) {
    const int gw = (int)((blockIdx.x * (size_t)blockDim.x + threadIdx.x) >> 5);
    if (gw >= E) return;
    const int lane = threadIdx.x & 31;
    const int r = erow[gw], c = ecol[gw];
    const float w = ew[gw];
    const float2 v = *(const float2*)(h + (size_t)c * OUT_C + lane * 2);
    float* __restrict__ dst = acc + (size_t)r * OUT_C + lane * 2;
    atomicAdd(dst + 0, w * v.x);
    atomicAdd(dst + 1, w * v.y);
}

// ---------------------------------------------------------------------------
// log_softmax over 64 classes, one wave32 per row, in place.
// ---------------------------------------------------------------------------
__global__ __launch_bounds__(256) void lsm64(float* __restrict__ out, int N) {
    const int row = (int)((blockIdx.x * (size_t)blockDim.x + threadIdx.x) >> 5);
    if (row >= N) return;
    const int lane = threadIdx.x & 31;
    float* p = out + (size_t)row * OUT_C;
    const float v0 = p[lane], v1 = p[lane + 32];
    float m = fmaxf(v0, v1);
#pragma unroll
    for (int o = 16; o; o >>= 1) m = fmaxf(m, __shfl_xor(m, o));
    float s = __expf(v0 - m) + __expf(v1 - m);
#pragma unroll
    for (int o = 16; o; o >>= 1) s += __shfl_xor(s, o);
    const float lse = m + __logf(s);
    p[lane]      = v0 - lse;
    p[lane + 32] = v1 - lse;
}

// ---------------------------------------------------------------------------
extern "C" void kernel_launch(void* const* d_in, const int* in_sizes, int n_in,
                              void* d_out, int out_size, void* d_ws, size_t ws_size,
                              hipStream_t stream) {
    const float* x    = (const float*)d_in[0];
    const int*   erow = (const int*)d_in[1];
    const int*   ecol = (const int*)d_in[2];
    const float* ew   = (const float*)d_in[3];
    const float* W1   = (const float*)d_in[4];
    const float* b1   = (const float*)d_in[5];
    const float* W2   = (const float*)d_in[6];
    const float* b2   = (const float*)d_in[7];
    float* out = (float*)d_out;

    const int N = in_sizes[0] / IN_C;
    const int E = in_sizes[1];

    float* h1   = (float*)d_ws;                 // N*256 f32
    float* acc1 = h1 + (size_t)N * HID_C;       // N*256 f32
    float* h2   = h1;                           // reuse dead h1 region: N*64 f32

    const int row32Blocks = (N + 31) / 32;
    const int row16Blocks = (N + 15) / 16;
    const int edgeBlocks  = (E + 7) / 8;         // 8 waves (edges) per 256-thread block

    // layer 1: linear -> spmm -> (relu fused into gemm2 A-read)
    zerok<<<2048, 256, 0, stream>>>(acc1, (size_t)N * HID_C);
    gemm1<<<row32Blocks, 256, 0, stream>>>(x, W1, b1, h1, N);
    spmm256<<<edgeBlocks, 256, 0, stream>>>(erow, ecol, ew, h1, acc1, E);

    // layer 2: linear -> spmm -> log_softmax
    gemm2<<<row16Blocks, 128, 0, stream>>>(acc1, W2, b2, h2, N);
    zerok<<<2048, 256, 0, stream>>>(out, (size_t)N * OUT_C);
    spmm64<<<edgeBlocks, 256, 0, stream>>>(erow, ecol, ew, h2, out, E);
    lsm64<<<(N + 7) / 8, 256, 0, stream>>>(out, N);
}